// PatchChamferDistance_62723702391633
// MI455X (gfx1250) — compile-verified
//
#include <hip/hip_runtime.h>

typedef __attribute__((ext_vector_type(2))) float v2f;
typedef __attribute__((ext_vector_type(8))) float v8f;

#define NPTS 256   // points per patch
#define TILE 16

// One workgroup (256 threads = 8 waves) per patch.
// Each wave computes two 16-row strips of the 256x256 dist^2 matrix.
// The entire dist^2 tile is produced by ONE V_WMMA_F32_16X16X4_F32:
//   A row m  = (-2x_m, -2y_m, -2z_m, 1)      (pred, scaled)
//   B col n  = ( x'_n,  y'_n,  z'_n, ||q_n||^2)
//   C[m][n]  = ||p_m||^2
//   => D[m][n] = ||p||^2 + ||q||^2 - 2 p.q
// The max(.,0) clamp commutes with min, so it is applied once after the
// min-reductions instead of per matrix element.
__global__ __launch_bounds__(256) void chamfer_patch_kernel(
    const float* __restrict__ pred,
    const float* __restrict__ tgt,
    float* __restrict__ per_patch)
{
    __shared__ float    sPred[NPTS * 4];   // (-2x, -2y, -2z, 1)
    __shared__ float    sTgt [NPTS * 4];   // ( x,   y,   z, tn)
    __shared__ float    sPn  [NPTS];       // ||pred||^2 (for C init)
    __shared__ unsigned sColMin[NPTS];     // float bits (values >= 0 only)
    __shared__ float    sRowMin[NPTS];
    __shared__ float    sPartial[8];

    const int tid   = threadIdx.x;
    const int patch = blockIdx.x;
    const long base = (long)patch * NPTS * 3;

    // ---- stage patch into LDS ----
    {
        float x = pred[base + tid * 3 + 0];
        float y = pred[base + tid * 3 + 1];
        float z = pred[base + tid * 3 + 2];
        sPred[tid * 4 + 0] = -2.0f * x;
        sPred[tid * 4 + 1] = -2.0f * y;
        sPred[tid * 4 + 2] = -2.0f * z;
        sPred[tid * 4 + 3] = 1.0f;
        sPn[tid] = x * x + y * y + z * z;

        x = tgt[base + tid * 3 + 0];
        y = tgt[base + tid * 3 + 1];
        z = tgt[base + tid * 3 + 2];
        sTgt[tid * 4 + 0] = x;
        sTgt[tid * 4 + 1] = y;
        sTgt[tid * 4 + 2] = z;
        sTgt[tid * 4 + 3] = x * x + y * y + z * z;   // ||q||^2

        sColMin[tid] = 0x7F800000u;   // +inf bits
    }
    __syncthreads();

    const int wave = tid >> 5;
    const int lane = tid & 31;
    const int half = lane >> 4;   // K/M group: 0 = low, 1 = high
    const int lh   = lane & 15;

    for (int s = 0; s < 2; ++s) {
        const int rb = (wave + s * 8) * TILE;   // row-strip base

        // A tile 16x4: lanes 0-15 hold K=0,1; lanes 16-31 hold K=2,3 (row M=lh)
        v2f a;
        a.x = sPred[(rb + lh) * 4 + 2 * half + 0];
        a.y = sPred[(rb + lh) * 4 + 2 * half + 1];

        // C init: element r corresponds to row M = r + 8*half
        v8f cinit;
#pragma unroll
        for (int r = 0; r < 8; ++r) cinit[r] = sPn[rb + r + 8 * half];

        float rowacc[8];
#pragma unroll
        for (int r = 0; r < 8; ++r) rowacc[r] = __builtin_inff();

        for (int cj = 0; cj < 16; ++cj) {
            const int cb = cj * TILE;

            // B tile 4x16 (mirror layout of A), col N = lh
            v2f b;
            b.x = sTgt[(cb + lh) * 4 + 2 * half + 0];
            b.y = sTgt[(cb + lh) * 4 + 2 * half + 1];

            // One WMMA yields the finished (unclamped) dist^2 tile
            v8f c = __builtin_amdgcn_wmma_f32_16x16x4_f32(
                        /*neg_a=*/false, a, /*neg_b=*/false, b,
                        /*c_mod=*/(short)0, cinit,
                        /*reuse_a=*/false, /*reuse_b=*/false);

            // running per-row (per-lane-column) minima — no clamp here
#pragma unroll
            for (int r = 0; r < 8; ++r)
                rowacc[r] = fminf(rowacc[r], c[r]);

            // column min over this lane's 8 rows: min3-friendly tree
            float m0 = fminf(fminf(c[0], c[1]), c[2]);
            float m1 = fminf(fminf(c[3], c[4]), c[5]);
            float m2 = fminf(c[6], c[7]);
            float colcand = fminf(fminf(m0, m1), m2);
            // clamp once, THEN the uint-ordered atomic (value is >= 0)
            colcand = fmaxf(colcand, 0.0f);
            // Both half-wave lanes min the same column slot; the DS atomic
            // merges the halves.
            atomicMin(&sColMin[cb + lh], __float_as_uint(colcand));
        }

        // row-min: reduce across the 16 lanes of each half, clamp once
#pragma unroll
        for (int r = 0; r < 8; ++r) {
            float v = rowacc[r];
            v = fminf(v, __shfl_xor(v, 1, 32));
            v = fminf(v, __shfl_xor(v, 2, 32));
            v = fminf(v, __shfl_xor(v, 4, 32));
            v = fminf(v, __shfl_xor(v, 8, 32));
            rowacc[r] = fmaxf(v, 0.0f);
        }
        if (lh == 0) {
#pragma unroll
            for (int r = 0; r < 8; ++r)
                sRowMin[rb + r + 8 * half] = rowacc[r];
        }
    }
    __syncthreads();

    // ---- per-patch: mean(rowmin) + mean(colmin) = sum(both)/256 ----
    float v = sRowMin[tid] + __uint_as_float(sColMin[tid]);
    v += __shfl_xor(v, 1, 32);
    v += __shfl_xor(v, 2, 32);
    v += __shfl_xor(v, 4, 32);
    v += __shfl_xor(v, 8, 32);
    v += __shfl_xor(v, 16, 32);
    if (lane == 0) sPartial[wave] = v;
    __syncthreads();
    if (tid == 0) {
        float t = 0.0f;
#pragma unroll
        for (int i = 0; i < 8; ++i) t += sPartial[i];
        per_patch[patch] = t * (1.0f / (float)NPTS);
    }
}

// Deterministic final mean over the per-patch values.
__global__ __launch_bounds__(256) void chamfer_mean_kernel(
    const float* __restrict__ per_patch, float* __restrict__ out, int n)
{
    __shared__ float sPartial[8];
    const int tid  = threadIdx.x;
    const int lane = tid & 31;
    const int wave = tid >> 5;

    float s = 0.0f;
    for (int i = tid; i < n; i += 256) s += per_patch[i];
    s += __shfl_xor(s, 1, 32);
    s += __shfl_xor(s, 2, 32);
    s += __shfl_xor(s, 4, 32);
    s += __shfl_xor(s, 8, 32);
    s += __shfl_xor(s, 16, 32);
    if (lane == 0) sPartial[wave] = s;
    __syncthreads();
    if (tid == 0) {
        float t = 0.0f;
#pragma unroll
        for (int i = 0; i < 8; ++i) t += sPartial[i];
        out[0] = t / (float)n;
    }
}

extern "C" void kernel_launch(void* const* d_in, const int* in_sizes, int n_in,
                              void* d_out, int out_size, void* d_ws, size_t ws_size,
                              hipStream_t stream)
{
    const float* pred = (const float*)d_in[0];
    const float* tgt  = (const float*)d_in[1];
    float* out = (float*)d_out;
    float* ws  = (float*)d_ws;

    const int BG = in_sizes[0] / (NPTS * 3);   // 32*64 = 2048 patches

    chamfer_patch_kernel<<<BG, 256, 0, stream>>>(pred, tgt, ws);
    chamfer_mean_kernel<<<1, 256, 0, stream>>>(ws, out, BG);
}